// Head_4483945857598
// MI455X (gfx1250) — compile-verified
//
#include <hip/hip_runtime.h>
#include <cstdint>

#define THREADS 1024
#define KDET    512
#define CAND    1024
#define NBINS   16384          // 64KB LDS histogram (reused for cand/box/area/keep)
#define CONF_T  0.3f
#define NMS_T   0.5f

__device__ __forceinline__ float sigmoidf_(float x) {
    return 1.0f / (1.0f + __expf(-x));
}

// ---- CDNA5 async global->LDS path (ASYNCcnt), CDNA5 ISA ----
__device__ __forceinline__ void async_g2l_b128(uint32_t lds_byte_addr, const float4* gptr) {
    // per-lane 16B copy: global vaddr -> LDS vdst; tracked by ASYNCcnt
    asm volatile("global_load_async_to_lds_b128 %0, %1, off"
                 :: "v"(lds_byte_addr), "v"(gptr)
                 : "memory");
}
__device__ __forceinline__ void wait_async0() { asm volatile("s_wait_asynccnt 0x0" ::: "memory"); }
__device__ __forceinline__ void wait_async1() { asm volatile("s_wait_asynccnt 0x1" ::: "memory"); }

__global__ __launch_bounds__(THREADS)
void head_kernel(const float* __restrict__ p8,
                 const float* __restrict__ p16,
                 const float* __restrict__ p32,
                 float* __restrict__ out)
{
    // ---- dynamic LDS: 64KB, phase-reused ----
    extern __shared__ unsigned char smem_raw[];
    unsigned*           hist  = (unsigned*)smem_raw;                 // [0,65536)   phase A/B
    unsigned long long* cand  = (unsigned long long*)smem_raw;       // [0,8192)    phase C+
    float4*             boxL  = (float4*)(smem_raw + 8192);          // [8192,16384)
    float*              areaL = (float*) (smem_raw + 16384);         // [16384,18432)
    int*                keepL = (int*)   (smem_raw + 18432);         // [18432,20480)

    __shared__ float4   stage[2][THREADS];   // async double-buffer staging (32KB)
    __shared__ unsigned csum[THREADS];
    __shared__ int      sThr;
    __shared__ unsigned sAbove, sCnt1, sCnt2;

    const int b   = blockIdx.x;
    const int tid = threadIdx.x;

    const float* layers[3] = { p8, p16, p32 };
    const int   HWs[3]  = { 25600, 6400, 1600 };
    const int   Ws[3]   = { 160, 80, 40 };
    const int   LOFF[3] = { 0, 76800, 96000 };
    const float STR[3]  = { 8.f, 16.f, 32.f };
    // ANCHORS / stride, from the reference
    const float AW[3][3] = {{1.25f, 2.0f,   4.125f  },
                            {1.875f,3.875f, 3.6875f },
                            {3.625f,4.875f, 11.65625f}};
    const float AH[3][3] = {{1.625f, 3.75f,   2.875f },
                            {3.8125f,2.8125f, 7.4375f},
                            {2.8125f,6.1875f, 10.1875f}};

    // ---------- Phase A: score histogram (async b128-staged stream) ----------
    for (int i = tid; i < NBINS; i += THREADS) hist[i] = 0u;
    __syncthreads();

    const uint32_t lds0 = (uint32_t)(uintptr_t)&stage[0][tid];  // flat low 32 bits = LDS byte addr
    const uint32_t lds1 = (uint32_t)(uintptr_t)&stage[1][tid];

    for (int l = 0; l < 3; ++l) {
        const int n4 = HWs[l] >> 2;            // HW divisible by 4 for all layers
        for (int a = 0; a < 3; ++a) {
            const float4* basep4 = (const float4*)(layers[l] +
                                   ((size_t)b * 18 + a * 6 + 4) * (size_t)HWs[l]); // obj channel
            int i = tid, buf = 0;
            if (i < n4) async_g2l_b128(lds0, basep4 + i);
            while (i < n4) {
                const int inx = i + THREADS;
                const bool more = (inx < n4);
                if (more) { async_g2l_b128(buf ? lds0 : lds1, basep4 + inx); wait_async1(); }
                else      { wait_async0(); }
                const float4 v = buf ? stage[1][tid] : stage[0][tid];
                const float sv[4] = { sigmoidf_(v.x), sigmoidf_(v.y),
                                      sigmoidf_(v.z), sigmoidf_(v.w) };
                #pragma unroll
                for (int c = 0; c < 4; ++c) {
                    if (sv[c] > CONF_T) {
                        int k = (int)(sv[c] * (float)NBINS);
                        if (k > NBINS - 1) k = NBINS - 1;
                        atomicAdd(&hist[k], 1u);
                    }
                }
                buf ^= 1; i = inx;
            }
        }
    }
    __syncthreads();

    // ---------- Phase B: find quantized rank-512 threshold (deterministic) ----------
    {
        unsigned ssum = 0;
        const int basebin = NBINS - 16 * (tid + 1);   // chunk tid covers 16 bins, top-down order
        #pragma unroll
        for (int t = 0; t < 16; ++t) ssum += hist[basebin + t];
        csum[tid] = ssum;
    }
    __syncthreads();
    if (tid == 0) {
        unsigned cum = 0; int c = 0;
        for (; c < THREADS; ++c) { if (cum + csum[c] >= KDET) break; cum += csum[c]; }
        int T = -1;
        if (c < THREADS) {
            const int hi = NBINS - 1 - 16 * c;
            for (int k = hi; k > hi - 16; --k) {
                const unsigned h = hist[k];
                if (cum + h >= KDET) { T = k; break; }
                cum += h;
            }
        }
        sThr = T; sAbove = cum; sCnt1 = 0u; sCnt2 = 0u;
    }
    __syncthreads();

    const int      T     = sThr;
    const unsigned above = sAbove;
    cand[tid] = 0ull;                 // overlays dead histogram region
    __syncthreads();

    // ---------- Phase C: compact candidates (second sweep hits L2, b128 loads) ----------
    for (int l = 0; l < 3; ++l) {
        const int n4 = HWs[l] >> 2;
        for (int a = 0; a < 3; ++a) {
            const float4* basep4 = (const float4*)(layers[l] +
                                   ((size_t)b * 18 + a * 6 + 4) * (size_t)HWs[l]);
            const int noff = LOFF[l] + a * HWs[l];
            for (int i = tid; i < n4; i += THREADS) {
                __builtin_prefetch(basep4 + i + 4 * THREADS, 0, 1);   // global_prefetch_b8
                const float4 v = basep4[i];
                const float sv[4] = { sigmoidf_(v.x), sigmoidf_(v.y),
                                      sigmoidf_(v.z), sigmoidf_(v.w) };
                #pragma unroll
                for (int c = 0; c < 4; ++c) {
                    const float s = sv[c];
                    if (s > CONF_T) {
                        int k = (int)(s * (float)NBINS);
                        if (k > NBINS - 1) k = NBINS - 1;
                        // key: score desc, then index asc (matches top_k tie order);
                        // unique low bits => deterministic final order
                        const unsigned long long key =
                            ((unsigned long long)__float_as_uint(s) << 32) |
                            (unsigned long long)(0xFFFFFFFFu - (unsigned)(noff + 4 * i + c));
                        if (k > T) {
                            const unsigned pos = atomicAdd(&sCnt1, 1u);  // pos < above < 512
                            if (pos < CAND) cand[pos] = key;
                        } else if (k == T) {
                            const unsigned pos = above + atomicAdd(&sCnt2, 1u);
                            if (pos < CAND) cand[pos] = key;
                        }
                    }
                }
            }
        }
    }
    __syncthreads();

    // ---------- Phase D: bitonic sort 1024 u64 keys, descending ----------
    for (int k = 2; k <= CAND; k <<= 1) {
        for (int j = k >> 1; j > 0; j >>= 1) {
            const int ixj = tid ^ j;
            if (ixj > tid) {
                const unsigned long long av = cand[tid], bv = cand[ixj];
                const bool desc = ((tid & k) == 0);
                if (desc ? (av < bv) : (av > bv)) { cand[tid] = bv; cand[ixj] = av; }
            }
            __syncthreads();
        }
    }

    // ---------- Phase E: decode boxes for top-512 (gather) ----------
    if (tid < KDET) {
        const unsigned long long key = cand[tid];
        const float sc = __uint_as_float((unsigned)(key >> 32));
        const bool valid = (sc > CONF_T);
        float4 bx = make_float4(0.f, 0.f, 0.f, 0.f);
        float ar = 0.f;
        if (valid) {
            const unsigned n = 0xFFFFFFFFu - (unsigned)(key & 0xFFFFFFFFull);
            int l, rem;
            if      (n < 76800u) { l = 0; rem = (int)n;          }
            else if (n < 96000u) { l = 1; rem = (int)n - 76800;  }
            else                 { l = 2; rem = (int)n - 96000;  }
            const int HW = HWs[l], W = Ws[l];
            const int a = rem / HW; const int p = rem - a * HW;
            const int y = p / W;    const int x = p - y * W;
            const float* cp = layers[l] + ((size_t)b * 18 + a * 6) * (size_t)HW + p;
            const float tx = sigmoidf_(cp[0]);
            const float ty = sigmoidf_(cp[(size_t)HW]);
            const float tw = sigmoidf_(cp[(size_t)2 * HW]);
            const float th = sigmoidf_(cp[(size_t)3 * HW]);
            const float st = STR[l];
            const float cx = (tx * 2.f - 0.5f + (float)x) * st;
            const float cy = (ty * 2.f - 0.5f + (float)y) * st;
            const float w  = (tw * 2.f) * (tw * 2.f) * AW[l][a] * st;
            const float h  = (th * 2.f) * (th * 2.f) * AH[l][a] * st;
            bx.x = cx - (w - 1.f) * 0.5f;
            bx.y = cy - (h - 1.f) * 0.5f;
            bx.z = cx + (w - 1.f) * 0.5f;
            bx.w = cy + (h - 1.f) * 0.5f;
            ar = (bx.z - bx.x) * (bx.w - bx.y);
        }
        boxL[tid] = bx; areaL[tid] = ar; keepL[tid] = valid ? 1 : 0;
    }
    __syncthreads();

    // ---------- Phase F: greedy NMS (512x512 in LDS) ----------
    for (int i = 0; i < KDET - 1; ++i) {
        if (keepL[i] && tid > i && tid < KDET && keepL[tid]) {
            const float4 bi = boxL[i];
            const float4 bj = boxL[tid];
            const float lx = fmaxf(bi.x, bj.x);
            const float ly = fmaxf(bi.y, bj.y);
            const float rx = fminf(bi.z, bj.z);
            const float ry = fminf(bi.w, bj.w);
            const float iw = fmaxf(rx - lx, 0.f);
            const float ih = fmaxf(ry - ly, 0.f);
            const float inter = iw * ih;
            const float iou = inter / (areaL[i] + areaL[tid] - inter + 1e-9f);
            if (iou > NMS_T) keepL[tid] = 0;
        }
        __syncthreads();
    }

    // ---------- Phase G: output [B,512,6] ----------
    if (tid < KDET) {
        const size_t o = (((size_t)b * KDET) + tid) * 6;
        const bool kp = keepL[tid] != 0;
        const float4 bx = boxL[tid];
        const float sc = __uint_as_float((unsigned)(cand[tid] >> 32));
        out[o + 0] = kp ? bx.x : 0.f;
        out[o + 1] = kp ? bx.y : 0.f;
        out[o + 2] = kp ? bx.z : 0.f;
        out[o + 3] = kp ? bx.w : 0.f;
        out[o + 4] = kp ? sc   : 0.f;
        out[o + 5] = 0.f;
    }
}

extern "C" void kernel_launch(void* const* d_in, const int* in_sizes, int n_in,
                              void* d_out, int out_size, void* d_ws, size_t ws_size,
                              hipStream_t stream) {
    (void)n_in; (void)out_size; (void)d_ws; (void)ws_size;
    const float* p8  = (const float*)d_in[0];
    const float* p16 = (const float*)d_in[1];
    const float* p32 = (const float*)d_in[2];
    float* out = (float*)d_out;
    const int B = in_sizes[0] / (18 * 160 * 160);
    head_kernel<<<dim3(B), dim3(THREADS), 64 * 1024, stream>>>(p8, p16, p32, out);
}